// InfGNNConv_83322365542770
// MI455X (gfx1250) — compile-verified
//
#include <hip/hip_runtime.h>

typedef __attribute__((ext_vector_type(2))) float v2f;
typedef __attribute__((ext_vector_type(8))) float v8f;

#define WPB 2  // waves (16-row tiles) per block

// ---------------------------------------------------------------------------
// One WMMA layer: D[16x64] = A[16x64] @ W^T[64x64] + C  (C preloaded in acc)
// A fragments from LDS tile T, B fragments (W[n][k]) from LDS weights.
// f32 16x16x4 layout: lane l -> row/col index (l&15); K pair (l<16 ? k,k+1 : k+2,k+3)
// ---------------------------------------------------------------------------
__device__ __forceinline__ void layer_wmma(const float* __restrict__ T,
                                           const float* __restrict__ W,
                                           v8f acc[4], int m, int lh) {
#pragma unroll
  for (int nt = 0; nt < 4; ++nt) {
    const int n = m + 16 * nt;
#pragma unroll
    for (int ks = 0; ks < 16; ++ks) {
      v2f a = *(const v2f*)(T + m * 64 + ks * 4 + lh * 2);
      v2f b = *(const v2f*)(W + n * 64 + ks * 4 + lh * 2);
      acc[nt] = __builtin_amdgcn_wmma_f32_16x16x4_f32(
          /*neg_a=*/false, a, /*neg_b=*/false, b,
          /*c_mod=*/(short)0, acc[nt], /*reuse_a=*/false, /*reuse_b=*/false);
    }
  }
}

// ---------------------------------------------------------------------------
// h = leaky_relu(leaky_relu(x + states@Ws^T) @ W1^T) @ W2^T   (fused, WMMA)
// ---------------------------------------------------------------------------
__global__ __launch_bounds__(WPB * 32) void node_mlp_kernel(
    const float* __restrict__ x, const float* __restrict__ states,
    const float* __restrict__ Ws, const float* __restrict__ W1,
    const float* __restrict__ W2, float* __restrict__ hout, int ntiles) {
  __shared__ float sWs[64 * 64];
  __shared__ float sW1[64 * 64];
  __shared__ float sW2[64 * 64];
  __shared__ float sT[WPB][16 * 64];

  const int tid = threadIdx.x;
  for (int i = tid; i < 64 * 64; i += WPB * 32) {
    sWs[i] = Ws[i];
    sW1[i] = W1[i];
    sW2[i] = W2[i];
  }
  __syncthreads();

  const int wave = tid >> 5;
  const int lane = tid & 31;
  const int m = lane & 15;   // row (A) / col (B,C) index within 16
  const int lh = lane >> 4;  // lane half selects K pair / row half of C
  const int tile = blockIdx.x * WPB + wave;

  if (tile < ntiles) {  // wave-uniform: EXEC stays all-ones for WMMA
    const int row0 = tile * 16;
    float* T = sT[wave];

    // stage states tile [16x64] into LDS (float4 per lane, 8 iterations)
#pragma unroll
    for (int j = 0; j < 8; ++j) {
      int idx4 = lane + 32 * j;
      int r = idx4 >> 4, c4 = idx4 & 15;
      *(float4*)(T + r * 64 + c4 * 4) =
          *(const float4*)(states + (size_t)(row0 + r) * 64 + c4 * 4);
    }
    asm volatile("s_wait_dscnt 0" ::: "memory");  // tile visible to all lanes

    // ---- layer 0: acc preloaded with x (fuses the residual add) ----
    v8f acc[4];
#pragma unroll
    for (int nt = 0; nt < 4; ++nt)
#pragma unroll
      for (int v = 0; v < 8; ++v)
        acc[nt][v] = x[(size_t)(row0 + v + lh * 8) * 64 + m + 16 * nt];
    layer_wmma(T, sWs, acc, m, lh);
#pragma unroll
    for (int nt = 0; nt < 4; ++nt)
#pragma unroll
      for (int v = 0; v < 8; ++v) {
        float t = acc[nt][v];
        acc[nt][v] = t >= 0.0f ? t : 0.01f * t;  // leaky relu
        sT[wave][(v + lh * 8) * 64 + m + 16 * nt] = acc[nt][v];
      }
    asm volatile("s_wait_dscnt 0" ::: "memory");

    // ---- layer 1 ----
#pragma unroll
    for (int nt = 0; nt < 4; ++nt) acc[nt] = (v8f)0.0f;
    layer_wmma(T, sW1, acc, m, lh);
#pragma unroll
    for (int nt = 0; nt < 4; ++nt)
#pragma unroll
      for (int v = 0; v < 8; ++v) {
        float t = acc[nt][v];
        acc[nt][v] = t >= 0.0f ? t : 0.01f * t;
        sT[wave][(v + lh * 8) * 64 + m + 16 * nt] = acc[nt][v];
      }
    asm volatile("s_wait_dscnt 0" ::: "memory");

    // ---- layer 2 (no activation) -> global h ----
#pragma unroll
    for (int nt = 0; nt < 4; ++nt) acc[nt] = (v8f)0.0f;
    layer_wmma(T, sW2, acc, m, lh);
#pragma unroll
    for (int nt = 0; nt < 4; ++nt)
#pragma unroll
      for (int v = 0; v < 8; ++v)
        hout[(size_t)(row0 + v + lh * 8) * 64 + m + 16 * nt] = acc[nt][v];
  }
}

// ---------------------------------------------------------------------------
__global__ void zero_kernel(float4* __restrict__ p, int n4) {
  int i = blockIdx.x * blockDim.x + threadIdx.x;
  if (i < n4) p[i] = make_float4(0.f, 0.f, 0.f, 0.f);
}

// msg = w[e] * h[src]; atomic scatter-add into agg[dst]. 32 lanes/edge, 2 dims/lane.
__global__ void edge_scatter_kernel(const int* __restrict__ ei,
                                    const float* __restrict__ w,
                                    const float* __restrict__ h,
                                    float* __restrict__ agg, int E) {
  int tid = blockIdx.x * blockDim.x + threadIdx.x;
  int e = tid >> 5;
  if (e >= E) return;
  int part = tid & 31;
  int dst = ei[e];      // edge_index[0][e] : segment (destination)
  int src = ei[E + e];  // edge_index[1][e] : gathered source
  float wv = w[e];
  float2 hv = *(const float2*)(h + (size_t)src * 64 + part * 2);
  float* o = agg + (size_t)dst * 64 + part * 2;
  unsafeAtomicAdd(o, wv * hv.x);      // global_atomic_add_f32
  unsafeAtomicAdd(o + 1, wv * hv.y);  // global_atomic_add_f32
}

// per-graph mean / unbiased std over [npg x 64] block of agg
__global__ void graph_stats_kernel(const float* __restrict__ agg,
                                   float* __restrict__ mu,
                                   float* __restrict__ sigma, int npg) {
  __shared__ float sh_s[256];
  __shared__ float sh_q[256];
  int g = blockIdx.x;
  int t = threadIdx.x;
  int d = t & 63;
  int r0 = t >> 6;  // 4 row-groups
  const float* base = agg + (size_t)g * npg * 64;
  float s = 0.f, q = 0.f;
  for (int r = r0; r < npg; r += 4) {
    float v = base[(size_t)r * 64 + d];
    s += v;
    q += v * v;
  }
  sh_s[t] = s;
  sh_q[t] = q;
  __syncthreads();
  if (t < 64) {
    s = sh_s[t] + sh_s[t + 64] + sh_s[t + 128] + sh_s[t + 192];
    q = sh_q[t] + sh_q[t + 64] + sh_q[t + 128] + sh_q[t + 192];
    float cnt = (float)npg;
    float mean = s / cnt;
    float var = (q - cnt * mean * mean) / fmaxf(cnt - 1.0f, 1.0f);
    var = fmaxf(var, 0.0f);
    mu[g * 64 + d] = mean;
    sigma[g * 64 + d] = sqrtf(var);
  }
}

// out = (agg - mu[g]) / (sigma[g] + eps) * gamma + beta   (in place on d_out)
__global__ void norm_kernel(float* __restrict__ out, const float* __restrict__ mu,
                            const float* __restrict__ sigma,
                            const float* __restrict__ gamma,
                            const float* __restrict__ beta,
                            const int* __restrict__ batch, int total) {
  int idx = blockIdx.x * blockDim.x + threadIdx.x;
  if (idx >= total) return;
  int n = idx >> 6, d = idx & 63;
  int g = batch[n];
  float a = out[idx];
  float sg = sigma[g * 64 + d];
  out[idx] = (a - mu[g * 64 + d]) / (sg + 1e-6f) * gamma[d] + beta[d];
}

// ---------------------------------------------------------------------------
extern "C" void kernel_launch(void* const* d_in, const int* in_sizes, int n_in,
                              void* d_out, int out_size, void* d_ws, size_t ws_size,
                              hipStream_t stream) {
  const float* x      = (const float*)d_in[0];
  const float* states = (const float*)d_in[1];
  const int*   ei     = (const int*)d_in[2];
  const float* w      = (const float*)d_in[3];
  const int*   batch  = (const int*)d_in[4];
  const float* Ws     = (const float*)d_in[6];
  const float* W1     = (const float*)d_in[7];
  const float* W2     = (const float*)d_in[8];
  const float* gamma  = (const float*)d_in[9];
  const float* beta   = (const float*)d_in[10];

  const int n_nodes = in_sizes[0] / 64;
  const int E = in_sizes[2] / 2;
  const int G = in_sizes[5];
  const int npg = n_nodes / G;

  float* out  = (float*)d_out;
  float* hbuf = (float*)d_ws;                      // [n_nodes*64]
  float* mu   = hbuf + (size_t)n_nodes * 64;       // [G*64]
  float* sig  = mu + (size_t)G * 64;               // [G*64]

  // 1) fused WMMA node MLP -> h (ws)
  const int ntiles = (n_nodes + 15) / 16;
  node_mlp_kernel<<<(ntiles + WPB - 1) / WPB, WPB * 32, 0, stream>>>(
      x, states, Ws, W1, W2, hbuf, ntiles);

  // 2) agg (= d_out) zero-init every call
  const int n4 = out_size / 4;
  zero_kernel<<<(n4 + 255) / 256, 256, 0, stream>>>((float4*)d_out, n4);

  // 3) edge gather + atomic scatter-add into agg
  const long long eth = (long long)E * 32;
  edge_scatter_kernel<<<(unsigned)((eth + 255) / 256), 256, 0, stream>>>(
      ei, w, hbuf, out, E);

  // 4) per-graph stats
  graph_stats_kernel<<<G, 256, 0, stream>>>(out, mu, sig, npg);

  // 5) GraphNorm in place
  const int total = n_nodes * 64;
  norm_kernel<<<(total + 255) / 256, 256, 0, stream>>>(out, mu, sig, gamma, beta,
                                                       batch, total);
}